// _BaseDimeNetExtractor_62079457296691
// MI455X (gfx1250) — compile-verified
//
#include <hip/hip_runtime.h>
#include <hip/hip_bf16.h>
#include <math.h>

typedef float v2f __attribute__((ext_vector_type(2)));
typedef float v8f __attribute__((ext_vector_type(8)));

#define HID      128
#define NRAD     6
#define TILE_E   128     // edges per macro-tile (per workgroup iteration)
#define SUB_E    16      // edges per wave sub-tile (WMMA M)
#define THREADS  256     // 8 wave32
#define HSTRIDE  130     // padded LDS row stride for h buffer (bank-conflict relief)

// LDS layout (float offsets)
#define OFF_W3   0
#define OFF_HB   (OFF_W3 + HID*HID)            // 16384
#define OFF_WRBF (OFF_HB + TILE_E*HSTRIDE)     // 33024
#define OFF_BRBF (OFF_WRBF + NRAD*HID)         // 33792
#define OFF_JID  (OFF_BRBF + HID)              // 33920
#define OFF_ZI   (OFF_JID + TILE_E)            // 34048
#define OFF_ZJ   (OFF_ZI + TILE_E)             // 34176
#define OFF_FREQ (OFF_ZJ + TILE_E)             // 34304
#define SMEM_FLOATS (OFF_FREQ + 8)             // 34312 floats = 137248 B (< 320 KB WGP LDS)

// fast silu: v * rcp(1+exp(-v)) -> v_exp/v_add/v_rcp/v_mul, no precise-div fixup chain
__device__ __forceinline__ float silu_f(float v) {
    return v * __builtin_amdgcn_rcpf(1.0f + __expf(-v));
}

// ---------------------------------------------------------------- zero init
__global__ void k_zero(float* __restrict__ out, float* __restrict__ cnt, int n_nodes) {
    long long t = (long long)blockIdx.x * blockDim.x + threadIdx.x;
    long long total = (long long)n_nodes * HID;
    if (t < total) out[t] = 0.0f;
    if (t < n_nodes) cnt[t] = 0.0f;
}

// ---------------------------------------------- precompute T1 = emb@W1 + b_lin,
// T2 = emb@W2  (95x128 each; tiny)
__global__ void k_pretab(const float* __restrict__ emb, const float* __restrict__ wlin,
                         const float* __restrict__ blin,
                         float* __restrict__ t1, float* __restrict__ t2) {
    int a = blockIdx.x;
    int n = threadIdx.x;
    float s1 = blin[n];
    float s2 = 0.0f;
    #pragma unroll 8
    for (int k = 0; k < HID; ++k) {
        float e = emb[a * HID + k];
        s1 = fmaf(e, wlin[k * HID + n], s1);
        s2 = fmaf(e, wlin[(HID + k) * HID + n], s2);
    }
    t1[a * HID + n] = s1;
    t2[a * HID + n] = s2;
}

// ---------------------------------------------------------------- main edge kernel
__global__ void __launch_bounds__(THREADS)
k_edges(const int* __restrict__ z, const float* __restrict__ pos,
        const int* __restrict__ eidx, int E, int ntiles,
        const float* __restrict__ freq, const float* __restrict__ wrbf,
        const float* __restrict__ brbf, const float* __restrict__ wlin,
        const float* __restrict__ t1, const float* __restrict__ t2,
        float* __restrict__ out, float* __restrict__ cnt)
{
    extern __shared__ float smem[];
    float* sW3 = smem + OFF_W3;
    float* sHB = smem + OFF_HB;
    float* sWR = smem + OFF_WRBF;
    float* sBR = smem + OFF_BRBF;
    int*   sJ  = (int*)(smem + OFF_JID);
    int*   sZI = (int*)(smem + OFF_ZI);
    int*   sZJ = (int*)(smem + OFF_ZJ);
    float* sFR = smem + OFF_FREQ;

    const int tid  = threadIdx.x;
    const float* W3 = wlin + 2 * HID * HID;

    // stage W3 and RBF-layer weights once per workgroup
    for (int k = tid; k < HID * HID; k += THREADS) sW3[k] = W3[k];
    for (int k = tid; k < NRAD * HID; k += THREADS) sWR[k] = wrbf[k];
    if (tid < HID)  sBR[tid] = brbf[tid];
    if (tid < NRAD) sFR[tid] = freq[tid];
    __syncthreads();

    const int lane = tid & 31;
    const int wave = tid >> 5;
    const int l16  = lane & 15;
    const int hi   = lane >> 4;   // 0: lanes 0-15, 1: lanes 16-31

    for (int tile = blockIdx.x; tile < ntiles; tile += gridDim.x) {
        const int ebase = tile * TILE_E;

        // ---- phase 1: per-edge RBF + h = silu(rbf@W_rbf + b_rbf) into LDS ----
        {
            const int el   = tid >> 1;   // 0..127 : edge within macro-tile
            const int half = tid & 1;    // 2 threads per edge, 64 cols each
            const int e  = ebase + el;
            const int vi = eidx[e];
            const int vj = eidx[E + e];
            float dx = pos[vi*3+0] - pos[vj*3+0];
            float dy = pos[vi*3+1] - pos[vj*3+1];
            float dz = pos[vi*3+2] - pos[vj*3+2];
            float d  = __builtin_amdgcn_sqrtf(dx*dx + dy*dy + dz*dz);
            float x  = d * 0.2f;               // d / CUTOFF
            float env = 0.0f;
            if (x < 1.0f) {                    // PyG Envelope, p=6
                float x2 = x * x;
                float x5 = x2 * x2 * x;
                env = __builtin_amdgcn_rcpf(x) + x5 * (-28.0f + x * (48.0f - 21.0f * x));
            }
            float rbf[NRAD];
            #pragma unroll
            for (int r = 0; r < NRAD; ++r) rbf[r] = env * __sinf(sFR[r] * x);

            if (half == 0) {
                sJ[el]  = vj;
                sZI[el] = z[vi];
                sZJ[el] = z[vj];
                atomicAdd(&cnt[vj], 1.0f);
            }
            const int c0 = half * 64;
            #pragma unroll 4
            for (int c = 0; c < 64; ++c) {
                int col = c0 + c;
                float acc = sBR[col];
                #pragma unroll
                for (int r = 0; r < NRAD; ++r)
                    acc = fmaf(rbf[r], sWR[r * HID + col], acc);
                sHB[el * HSTRIDE + col] = silu_f(acc);
            }
        }
        __syncthreads();

        // ---- phase 2: per-wave 16-edge sub-tile, fp32 WMMA h@W3 ----
        {
            const int est = wave * SUB_E;
            v8f acc[8];
            // accumulator init from precomputed tables: T1[z_i] + T2[z_j]
            // C/D layout: VGPR r holds M=r (lanes 0-15) and M=r+8 (lanes 16-31), N=lane&15
            #pragma unroll
            for (int nt = 0; nt < 8; ++nt) {
                #pragma unroll
                for (int r = 0; r < 8; ++r) {
                    int m  = r + hi * 8;
                    int n  = nt * 16 + l16;
                    int el = est + m;
                    int zi = sZI[el];
                    int zj = sZJ[el];
                    acc[nt][r] = t1[zi * HID + n] + t2[zj * HID + n];
                }
            }
            // K loop: 32 steps of K=4 over HID=128
            #pragma unroll 1
            for (int kt = 0; kt < 32; ++kt) {
                const int kb = kt * 4;
                // A (16x4): lane m (0-15) holds K=kb+0,kb+1; lane m+16 holds K=kb+2,kb+3
                v2f a = *(const v2f*)(&sHB[(est + l16) * HSTRIDE + kb + hi * 2]);
                #pragma unroll
                for (int nt = 0; nt < 8; ++nt) {
                    // B (4x16): vgpr0 = row kb+0 (lanes 0-15) / kb+2 (lanes 16-31); vgpr1 = +1
                    v2f b;
                    b.x = sW3[(kb + hi * 2    ) * HID + nt * 16 + l16];
                    b.y = sW3[(kb + hi * 2 + 1) * HID + nt * 16 + l16];
                    acc[nt] = __builtin_amdgcn_wmma_f32_16x16x4_f32(
                        false, a, false, b, (short)0, acc[nt], false, false);
                }
            }
            // epilogue: silu + scatter-add into out[j]
            #pragma unroll
            for (int nt = 0; nt < 8; ++nt) {
                #pragma unroll
                for (int r = 0; r < 8; ++r) {
                    int m  = r + hi * 8;
                    int n  = nt * 16 + l16;
                    int vj = sJ[est + m];
                    atomicAdd(&out[vj * HID + n], silu_f(acc[nt][r]));
                }
            }
        }
        __syncthreads();
    }
}

// ------------------------------------------------ scalar remainder (E % 128 edges)
__global__ void k_edges_rem(const int* __restrict__ z, const float* __restrict__ pos,
                            const int* __restrict__ eidx, int E, int start,
                            const float* __restrict__ freq, const float* __restrict__ wrbf,
                            const float* __restrict__ brbf, const float* __restrict__ wlin,
                            const float* __restrict__ t1, const float* __restrict__ t2,
                            float* __restrict__ out, float* __restrict__ cnt)
{
    int e = start + blockIdx.x * blockDim.x + threadIdx.x;
    if (e >= E) return;
    const float* W3 = wlin + 2 * HID * HID;
    int vi = eidx[e];
    int vj = eidx[E + e];
    float dx = pos[vi*3+0] - pos[vj*3+0];
    float dy = pos[vi*3+1] - pos[vj*3+1];
    float dz = pos[vi*3+2] - pos[vj*3+2];
    float d  = __builtin_amdgcn_sqrtf(dx*dx + dy*dy + dz*dz);
    float x  = d * 0.2f;
    float env = 0.0f;
    if (x < 1.0f) {
        float x2 = x * x;
        float x5 = x2 * x2 * x;
        env = __builtin_amdgcn_rcpf(x) + x5 * (-28.0f + x * (48.0f - 21.0f * x));
    }
    float rbf[NRAD];
    for (int r = 0; r < NRAD; ++r) rbf[r] = env * __sinf(freq[r] * x);
    float h[HID];
    for (int c = 0; c < HID; ++c) {
        float a = brbf[c];
        for (int r = 0; r < NRAD; ++r) a = fmaf(rbf[r], wrbf[r * HID + c], a);
        h[c] = silu_f(a);
    }
    atomicAdd(&cnt[vj], 1.0f);
    int zi = z[vi], zj = z[vj];
    for (int n = 0; n < HID; ++n) {
        float a = t1[zi * HID + n] + t2[zj * HID + n];
        for (int k = 0; k < HID; ++k) a = fmaf(h[k], W3[k * HID + n], a);
        atomicAdd(&out[vj * HID + n], silu_f(a));
    }
}

// ---------------------------------------------------------------- mean finalize
__global__ void k_fin(float* __restrict__ out, const float* __restrict__ cnt, int n_nodes) {
    long long t = (long long)blockIdx.x * blockDim.x + threadIdx.x;
    if (t < (long long)n_nodes * HID) {
        float c = cnt[t >> 7];
        out[t] = out[t] * __builtin_amdgcn_rcpf(fmaxf(c, 1.0f));
    }
}

extern "C" void kernel_launch(void* const* d_in, const int* in_sizes, int n_in,
                              void* d_out, int out_size, void* d_ws, size_t ws_size,
                              hipStream_t stream) {
    const int*   z    = (const int*)  d_in[0];
    const float* pos  = (const float*)d_in[1];
    const int*   eidx = (const int*)  d_in[2];
    const float* freq = (const float*)d_in[3];
    const float* emb  = (const float*)d_in[4];
    const float* wrbf = (const float*)d_in[5];
    const float* brbf = (const float*)d_in[6];
    const float* wlin = (const float*)d_in[7];
    const float* blin = (const float*)d_in[8];
    float* out = (float*)d_out;

    const int N    = in_sizes[0];
    const int E    = in_sizes[2] / 2;
    const int MAXZ = in_sizes[4] / HID;

    float* t1  = (float*)d_ws;
    float* t2  = t1 + (size_t)MAXZ * HID;
    float* cnt = t2 + (size_t)MAXZ * HID;

    const long long total = (long long)N * HID;
    const int zb = (int)((total + 255) / 256);

    k_zero<<<zb, 256, 0, stream>>>(out, cnt, N);
    k_pretab<<<MAXZ, HID, 0, stream>>>(emb, wlin, blin, t1, t2);

    const int ntiles = E / TILE_E;
    if (ntiles > 0) {
        int grid = ntiles < 1280 ? ntiles : 1280;
        size_t smem = (size_t)SMEM_FLOATS * sizeof(float);
        k_edges<<<grid, THREADS, smem, stream>>>(z, pos, eidx, E, ntiles,
                                                 freq, wrbf, brbf, wlin, t1, t2, out, cnt);
    }
    const int rem = E - ntiles * TILE_E;
    if (rem > 0) {
        k_edges_rem<<<(rem + 127) / 128, 128, 0, stream>>>(z, pos, eidx, E, ntiles * TILE_E,
                                                           freq, wrbf, brbf, wlin, t1, t2, out, cnt);
    }
    k_fin<<<zb, 256, 0, stream>>>(out, cnt, N);
}